// NormalizedEmbedding_60447369724215
// MI455X (gfx1250) — compile-verified
//
#include <hip/hip_runtime.h>
#include <stdint.h>
#include <math.h>

// NormalizedEmbedding on gfx1250 (MI455X):
//   out[t, :] = weight[idx[t], :] / max(||weight[idx[t], :]||_2, 1e-12)
// (EMBED_SCALE * sqrt(DIM) == 1.0 exactly, so no extra rescale.)
//
// Bandwidth-bound gather+normalize (~268 MB traffic -> ~11.5 us at 23.3 TB/s).
// One wave32 per token row; rows staged through LDS with the CDNA5 async copy
// engine (GLOBAL_LOAD_ASYNC_TO_LDS_B128 / ASYNCcnt), double-buffered so the
// random-gather HBM latency is hidden. Row readback uses native DS loads
// (addrspace(3) pointers -> ds_load_b128), output uses non-temporal B128
// stores so L2 stays free for repeated weight rows.

#define NE_DIM        1024          // floats per row
#define NE_ROW_BYTES  4096
#define NE_WAVES_PB   8             // waves per block (256 threads, wave32)
#define NE_THREADS    (NE_WAVES_PB * 32)
#define NE_MAX_BLOCKS 1024

typedef float v4f __attribute__((ext_vector_type(4)));
typedef __attribute__((address_space(3))) const v4f* lds_v4f_cp;

// Workgroup-relative LDS byte offset of a __shared__ object (generic -> as3 -> int).
__device__ __forceinline__ unsigned ne_lds_off(const void* p) {
  return (unsigned)(uintptr_t)(__attribute__((address_space(3))) const void*)p;
}

// Stage one 4KB row into LDS: 8 async B128 copies, 512B (32 lanes x 16B) each.
// INST_OFFSET is added to BOTH the LDS and the global address (ISA 08 §4.4),
// so the per-chunk offset rides in the immediate; the per-lane 16B offset is
// pre-folded into both the LDS-address VGPR and the global-address VGPR pair.
__device__ __forceinline__ void ne_async_row(unsigned lds_lane, const float* g_lane) {
#define NE_CP(OFF)                                                        \
  asm volatile("global_load_async_to_lds_b128 %0, %1, off offset:" #OFF   \
               :: "v"(lds_lane), "v"(g_lane) : "memory")
  NE_CP(0);    NE_CP(512);  NE_CP(1024); NE_CP(1536);
  NE_CP(2048); NE_CP(2560); NE_CP(3072); NE_CP(3584);
#undef NE_CP
}

__global__ __launch_bounds__(NE_THREADS)
void NormalizedEmbedding_60447369724215_kernel(const int* __restrict__ idx,
                                               const float* __restrict__ w,
                                               float* __restrict__ out,
                                               int ntok) {
  // 2 row-buffers per wave: 8 waves * 2 * 4KB = 64KB (<= 320KB/WGP)
  __shared__ __align__(16) float lds[NE_WAVES_PB * 2 * NE_DIM];

  const int lane   = threadIdx.x & 31;
  const int waveIB = threadIdx.x >> 5;
  const int wave   = blockIdx.x * NE_WAVES_PB + waveIB;
  const int nwaves = gridDim.x * NE_WAVES_PB;

  // Per-buffer LDS byte addresses (lane's 16B slot), used for BOTH the async
  // copy destination and the ds_load readback (as addrspace(3) pointers).
  const unsigned laneB = (unsigned)(lane * 16);
  unsigned ldsb[2] = {
    ne_lds_off(&lds[(waveIB * 2 + 0) * NE_DIM]) + laneB,
    ne_lds_off(&lds[(waveIB * 2 + 1) * NE_DIM]) + laneB
  };

  int row = wave;
  if (row >= ntok) return;

  // Prime the pipeline: prefetch first row into buffer 0.
  ne_async_row(ldsb[0], w + (size_t)idx[row] * NE_DIM + lane * 4);

  int cur = 0;
  for (;;) {
    const int  nextRow = row + nwaves;
    const bool hasNext = nextRow < ntok;
    // Prefetch next row into the other buffer (dummy re-prefetch of the
    // current row on the last iteration keeps the ASYNCcnt discipline uniform).
    const int pf = hasNext ? nextRow : row;
    ne_async_row(ldsb[cur ^ 1], w + (size_t)idx[pf] * NE_DIM + lane * 4);

    // 16 async ops outstanding; waiting <=8 guarantees the 8 oldest (the
    // current row's data) have landed in LDS (async ops complete in order).
    asm volatile("s_wait_asynccnt 0x8" ::: "memory");

    // Read the row back with native DS loads: 32 floats/lane as 8x ds_load_b128
    // (addrspace(3) pointer -> immediate chunk offsets, no flat path).
    lds_v4f_cp lb = (lds_v4f_cp)ldsb[cur];
    v4f v[8];
    float s = 0.0f;
#pragma unroll
    for (int c = 0; c < 8; ++c) {
      v[c] = lb[c * 32];                 // c*512 bytes from this lane's slot
      s += v[c].x * v[c].x;
      s += v[c].y * v[c].y;
      s += v[c].z * v[c].z;
      s += v[c].w * v[c].w;
    }
    // wave32 butterfly reduction of the sum of squares
#pragma unroll
    for (int m = 16; m >= 1; m >>= 1) s += __shfl_xor(s, m, 32);

    const float inv = 1.0f / fmaxf(sqrtf(s), 1e-12f);

    // Streamed output: non-temporal B128 stores (written once, never re-read).
    v4f* op = (v4f*)(out + (size_t)row * NE_DIM) + lane;
#pragma unroll
    for (int c = 0; c < 8; ++c) {
      v4f r = v[c];
      r.x *= inv; r.y *= inv; r.z *= inv; r.w *= inv;
      __builtin_nontemporal_store(r, op + c * 32);
    }

    if (!hasNext) break;
    row = nextRow;
    cur ^= 1;
  }
}

extern "C" void kernel_launch(void* const* d_in, const int* in_sizes, int n_in,
                              void* d_out, int out_size, void* d_ws, size_t ws_size,
                              hipStream_t stream) {
  (void)n_in; (void)out_size; (void)d_ws; (void)ws_size;
  const int*   idx = (const int*)d_in[0];   // input: (8, 4096) token ids
  const float* w   = (const float*)d_in[1]; // weight: (128000, 1024) f32
  float*       out = (float*)d_out;         // (8, 4096, 1024) f32

  const int ntok   = in_sizes[0];           // 32768
  int blocks = (ntok + NE_WAVES_PB - 1) / NE_WAVES_PB;
  if (blocks > NE_MAX_BLOCKS) blocks = NE_MAX_BLOCKS;
  if (blocks < 1) blocks = 1;

  hipLaunchKernelGGL(NormalizedEmbedding_60447369724215_kernel,
                     dim3(blocks), dim3(NE_THREADS), 0, stream,
                     idx, w, out, ntok);
}